// ALCOVE_RBF_20177756357006
// MI455X (gfx1250) — compile-verified
//
#include <hip/hip_runtime.h>

#ifndef __has_builtin
#define __has_builtin(x) 0
#endif

#define TILE 128   // output tile edge (batch and exemplar)
#define KDIM 128   // feature dimension (fixed by the reference)
#define TPB  256   // 8 waves (wave32)

typedef __attribute__((address_space(1))) int* gint_p;
typedef __attribute__((address_space(3))) int* lint_p;

// ---- CDNA5 async global->LDS copy (per-lane LDS address => free transpose) ----
__device__ __forceinline__ void async_g2lds_b32(const float* g, float* l) {
#if __has_builtin(__builtin_amdgcn_global_load_async_to_lds_b32)
  __builtin_amdgcn_global_load_async_to_lds_b32(
      (gint_p)const_cast<float*>(g),
      (lint_p)l,
      /*imm offset*/ 0, /*cpol*/ 0);
#else
  *l = *g;  // synchronous fallback
#endif
}

__device__ __forceinline__ void wait_async0() {
#if __has_builtin(__builtin_amdgcn_s_wait_asynccnt)
  __builtin_amdgcn_s_wait_asynccnt(0);
#else
  asm volatile("s_wait_asynccnt 0" ::: "memory");
#endif
}

__global__ __launch_bounds__(TPB) void alcove_rbf_kernel(
    const float* __restrict__ X,     // [batch][KDIM] inputs
    const float* __restrict__ E,     // [ne][KDIM]    exemplars
    const float* __restrict__ attn,  // [KDIM]
    float* __restrict__ out,         // [batch][ne]
    int batch, int ne) {
  // d-major (transposed) tiles: sE[d][e], sX[d][b]
  __shared__ __align__(16) float sE[KDIM * TILE];
  __shared__ __align__(16) float sX[KDIM * TILE];

  const int tid   = threadIdx.x;
  const int eBase = blockIdx.x * TILE;
  const int bBase = blockIdx.y * TILE;

  // ---- Stage both tiles into LDS, transposing to d-major via async scatter.
  {
    const int elem = tid & (TILE - 1);  // which row of the tile this lane feeds
    const int d0   = tid >> 7;          // 0 or 1
    const float* ge = E + (size_t)(eBase + elem) * KDIM + d0;
    const float* gx = X + (size_t)(bBase + elem) * KDIM + d0;
    float* le = sE + d0 * TILE + elem;
    float* lx = sX + d0 * TILE + elem;
#pragma unroll
    for (int k = 0; k < KDIM / 2; ++k) {
      async_g2lds_b32(ge + 2 * k, le + 2 * k * TILE);
      async_g2lds_b32(gx + 2 * k, lx + 2 * k * TILE);
    }
  }
  wait_async0();
  __syncthreads();

  // ---- Fold attn into both tiles in LDS:  w_d*|e-x| == |w_d*e - w_d*x| (attn>0).
  {
    float4* sE4 = (float4*)sE;
    float4* sX4 = (float4*)sX;
    const int d0 = tid >> 5;  // 0..7 (wave-uniform)
    const int c  = tid & 31;  // float4 column within the d-row
#pragma unroll
    for (int k = 0; k < 16; ++k) {
      const int d   = d0 + 8 * k;
      const float w = attn[d];
      float4 ve = sE4[d * (TILE / 4) + c];
      float4 vx = sX4[d * (TILE / 4) + c];
      ve.x *= w; ve.y *= w; ve.z *= w; ve.w *= w;
      vx.x *= w; vx.y *= w; vx.z *= w; vx.w *= w;
      sE4[d * (TILE / 4) + c] = ve;
      sX4[d * (TILE / 4) + c] = vx;
    }
  }
  __syncthreads();

  // ---- 8x8 register tile per thread: 2 VALU ops (sub + add|abs|) per (pair,d).
  const int tx = tid & 15;  // exemplar group
  const int ty = tid >> 4;  // batch group
  float acc[8][8];
#pragma unroll
  for (int i = 0; i < 8; ++i)
#pragma unroll
    for (int j = 0; j < 8; ++j) acc[i][j] = 0.0f;

  const float4* cE4 = (const float4*)sE;
  const float4* cX4 = (const float4*)sX;
#pragma unroll 4
  for (int d = 0; d < KDIM; ++d) {
    const float4 e0 = cE4[d * (TILE / 4) + tx * 2 + 0];
    const float4 e1 = cE4[d * (TILE / 4) + tx * 2 + 1];
    const float4 x0 = cX4[d * (TILE / 4) + ty * 2 + 0];
    const float4 x1 = cX4[d * (TILE / 4) + ty * 2 + 1];
    const float ev[8] = {e0.x, e0.y, e0.z, e0.w, e1.x, e1.y, e1.z, e1.w};
    const float xv[8] = {x0.x, x0.y, x0.z, x0.w, x1.x, x1.y, x1.z, x1.w};
#pragma unroll
    for (int i = 0; i < 8; ++i) {
      const float xi = xv[i];
#pragma unroll
      for (int j = 0; j < 8; ++j) {
        acc[i][j] += __builtin_fabsf(ev[j] - xi);  // v_sub + v_add |src|
      }
    }
  }

  // ---- out = exp(-C*dist) = exp2(dist * (-C*log2(e)))  -> v_mul + v_exp_f32
  constexpr float kScale = -6.5f * 1.4426950408889634f;
  const int col = eBase + tx * 8;
#pragma unroll
  for (int i = 0; i < 8; ++i) {
    const int row = bBase + ty * 8 + i;
    float4 o0, o1;
    o0.x = __builtin_amdgcn_exp2f(acc[i][0] * kScale);
    o0.y = __builtin_amdgcn_exp2f(acc[i][1] * kScale);
    o0.z = __builtin_amdgcn_exp2f(acc[i][2] * kScale);
    o0.w = __builtin_amdgcn_exp2f(acc[i][3] * kScale);
    o1.x = __builtin_amdgcn_exp2f(acc[i][4] * kScale);
    o1.y = __builtin_amdgcn_exp2f(acc[i][5] * kScale);
    o1.z = __builtin_amdgcn_exp2f(acc[i][6] * kScale);
    o1.w = __builtin_amdgcn_exp2f(acc[i][7] * kScale);
    float4* p = (float4*)(out + (size_t)row * ne + col);
    p[0] = o0;
    p[1] = o1;
  }
}

extern "C" void kernel_launch(void* const* d_in, const int* in_sizes, int n_in,
                              void* d_out, int out_size, void* d_ws, size_t ws_size,
                              hipStream_t stream) {
  (void)n_in; (void)d_ws; (void)ws_size; (void)out_size;
  const float* X    = (const float*)d_in[0];  // inputs    [batch*128]
  const float* E    = (const float*)d_in[1];  // exemplars [ne*128]
  const float* attn = (const float*)d_in[2];  // attn      [128]
  float* out = (float*)d_out;

  const int ndims = in_sizes[2];          // 128
  const int batch = in_sizes[0] / ndims;  // 2048
  const int ne    = in_sizes[1] / ndims;  // 2048

  dim3 grid(ne / TILE, batch / TILE);
  alcove_rbf_kernel<<<grid, TPB, 0, stream>>>(X, E, attn, out, batch, ne);
}